// DiffPoolGNN_41979010351134
// MI455X (gfx1250) — compile-verified
//
#include <hip/hip_runtime.h>

typedef float v2f __attribute__((ext_vector_type(2)));
typedef float v8f __attribute__((ext_vector_type(8)));

#define CH 128

// ---------------- degree / normalization ----------------
__global__ __launch_bounds__(256) void k_init_deg(float* deg, int n) {
  int i = blockIdx.x * 256 + threadIdx.x;
  if (i < n) deg[i] = 1.0f;  // +1 self loop
}

__global__ __launch_bounds__(256) void k_count_deg(const int* __restrict__ dst,
                                                   float* deg, int e) {
  int i = blockIdx.x * 256 + threadIdx.x;
  if (i < e) atomicAdd(&deg[dst[i]], 1.0f);
}

__global__ __launch_bounds__(256) void k_rsqrt(float* deg, int n) {
  int i = blockIdx.x * 256 + threadIdx.x;
  if (i < n) deg[i] = rsqrtf(deg[i]);
}

// ---------------- WMMA f32 GEMM: H[n,128] = act(X)[n,128] @ W[128,128] ----------------
// block = 256 threads = 8 waves; block covers 16 rows, wave w covers cols [16w,16w+16)
// RELU templated so the inner loop is branch/select-free.
template <bool RELU>
__global__ __launch_bounds__(256) void k_gemm128_wmma(
    const float* __restrict__ X, const float* __restrict__ W,
    float* __restrict__ H, int nRows) {
  __shared__ float sWt[CH * CH];  // W transposed: sWt[col*128 + k]
  const int tid = threadIdx.x;
  for (int i = tid; i < CH * CH; i += 256) {
    int k = i >> 7, c = i & 127;
    sWt[c * CH + k] = W[i];
  }
  __syncthreads();

  const int lane  = tid & 31;
  const int wave  = tid >> 5;
  const int col0  = wave * 16;
  const int m     = lane & 15;           // row (A) / col (B,D) within tile
  const int khalf = (lane >> 4) << 1;    // K offset pair: 0 or 2
  const int row0  = blockIdx.x * 16;

  int r = row0 + m;
  int rclamp = (r < nRows) ? r : (nRows - 1);  // keep EXEC uniform for WMMA
  const float* __restrict__ xrow = X + (long)rclamp * CH + khalf;
  const float* __restrict__ wcol = sWt + (col0 + m) * CH + khalf;

  v8f acc = {0.f, 0.f, 0.f, 0.f, 0.f, 0.f, 0.f, 0.f};
#pragma unroll
  for (int k = 0; k < CH; k += 4) {
    float ax = xrow[k];
    float ay = xrow[k + 1];
    if (RELU) {                       // compile-time; single v_max_num_f32 each
      ax = (ax > 0.f) ? ax : 0.f;
      ay = (ay > 0.f) ? ay : 0.f;
    }
    v2f a; a.x = ax;        a.y = ay;
    v2f b; b.x = wcol[k];   b.y = wcol[k + 1];
    acc = __builtin_amdgcn_wmma_f32_16x16x4_f32(
        /*neg_a=*/false, a, /*neg_b=*/false, b,
        /*c_mod=*/(short)0, acc, /*reuse_a=*/false, /*reuse_b=*/false);
  }

  const int hi = (lane >> 4) * 8;  // lanes 0-15: rows v; lanes 16-31: rows v+8
#pragma unroll
  for (int v = 0; v < 8; ++v) {
    int rr = row0 + hi + v;
    if (rr < nRows) H[(long)rr * CH + col0 + m] = acc[v];
  }
}

// ---------------- out = h * dinv^2 + bias (self-loop term + bias init) ----------------
__global__ __launch_bounds__(256) void k_self_bias(
    const float* __restrict__ h, const float* __restrict__ dinv,
    const float* __restrict__ bias, float* __restrict__ out, long total) {
  long t = (long)blockIdx.x * 256 + threadIdx.x;
  if (t >= total) return;
  int i = (int)(t >> 7);
  int c = (int)(t & 127);
  float di = dinv[i];
  out[t] = h[t] * di * di + bias[c];
}

// ---------------- edge scatter: out[dst] += h[src] * dinv[src]*dinv[dst] ----------------
// one wave per edge; lane handles 4 channels (float4 gather, 4 atomic f32 adds)
__global__ __launch_bounds__(256) void k_edge_agg(
    const float* __restrict__ h, const int* __restrict__ src,
    const int* __restrict__ dst, const float* __restrict__ dinv,
    float* __restrict__ out, int e) {
  int t = blockIdx.x * 256 + threadIdx.x;
  int eid = t >> 5;
  if (eid >= e) return;
  int lane = t & 31;
  int s = src[eid], d = dst[eid];
  float norm = dinv[s] * dinv[d];
  int c = lane * 4;
  const float4 v = *(const float4*)(h + (long)s * CH + c);
  float* o = out + (long)d * CH + c;
  atomicAdd(o + 0, v.x * norm);
  atomicAdd(o + 1, v.y * norm);
  atomicAdd(o + 2, v.z * norm);
  atomicAdd(o + 3, v.w * norm);
}

// ---------------- pooling ----------------
__global__ __launch_bounds__(256) void k_zero(float* p, int n) {
  int i = blockIdx.x * 256 + threadIdx.x;
  if (i < n) p[i] = 0.0f;
}

// one wave per node; relu fused into the read
__global__ __launch_bounds__(256) void k_pool(
    const float* __restrict__ x, const int* __restrict__ batch,
    float* __restrict__ pooled, float* __restrict__ counts, int n) {
  int t = blockIdx.x * 256 + threadIdx.x;
  int node = t >> 5;
  if (node >= n) return;
  int lane = t & 31;
  int g = batch[node];
  int c = lane * 4;
  const float4 v = *(const float4*)(x + (long)node * CH + c);
  float* p = pooled + (long)g * CH + c;
  atomicAdd(p + 0, (v.x > 0.f) ? v.x : 0.f);
  atomicAdd(p + 1, (v.y > 0.f) ? v.y : 0.f);
  atomicAdd(p + 2, (v.z > 0.f) ? v.z : 0.f);
  atomicAdd(p + 3, (v.w > 0.f) ? v.w : 0.f);
  if (lane == 0) atomicAdd(&counts[g], 1.0f);
}

// ---------------- final linear: out[g, :cout] = (pooled[g]/cnt) @ Wlin + blin ----------------
__global__ __launch_bounds__(128) void k_final(
    const float* __restrict__ pooled, const float* __restrict__ counts,
    const float* __restrict__ Wlin, const float* __restrict__ blin,
    float* __restrict__ out, int cout) {
  __shared__ float sv[CH];
  int g = blockIdx.x;
  int c = threadIdx.x;
  float cnt = counts[g];
  cnt = (cnt > 1.0f) ? cnt : 1.0f;
  sv[c] = pooled[(long)g * CH + c] / cnt;
  __syncthreads();
  if (c < cout) {
    float acc = blin[c];
#pragma unroll 8
    for (int k = 0; k < CH; ++k) acc += sv[k] * Wlin[k * cout + c];
    out[g * cout + c] = acc;
  }
}

extern "C" void kernel_launch(void* const* d_in, const int* in_sizes, int n_in,
                              void* d_out, int out_size, void* d_ws, size_t ws_size,
                              hipStream_t stream) {
  const float* x    = (const float*)d_in[0];
  const int*   ei   = (const int*)d_in[1];
  const int*   batch= (const int*)d_in[2];
  // d_in[3],[4] = W1p,b1p  (dead: feeds only the unused softmax branch)
  const float* W1e  = (const float*)d_in[5];
  const float* b1e  = (const float*)d_in[6];
  // d_in[7],[8] = W2p,b2p  (dead)
  const float* W2e  = (const float*)d_in[9];
  const float* b2e  = (const float*)d_in[10];
  const float* Wlin = (const float*)d_in[11];
  const float* blin = (const float*)d_in[12];
  float* out = (float*)d_out;

  const int N    = in_sizes[0] / CH;
  const int E    = in_sizes[1] / 2;
  const int COUT = in_sizes[12];
  const int G    = out_size / COUT;
  const int* src = ei;
  const int* dstv = ei + E;

  float* ws = (float*)d_ws;
  long Npad = ((long)N + 3) & ~3L;           // keep float4 alignment downstream
  float* dinv   = ws;
  float* hbuf   = ws + Npad;                 // N*128
  float* gbuf   = hbuf + (long)N * CH;       // N*128
  float* pooled = gbuf + (long)N * CH;       // G*128
  float* counts = pooled + (long)G * CH;     // G (contiguous after pooled)

  dim3 blk(256);
  const long tot = (long)N * CH;
  const int gemmBlocks = (N + 15) / 16;
  const int edgeBlocks = (int)(((long)E * 32 + 255) / 256);
  const int poolBlocks = (int)(((long)N * 32 + 255) / 256);

  // degree -> dinv (shared by both layers)
  k_init_deg<<<(N + 255) / 256, blk, 0, stream>>>(dinv, N);
  k_count_deg<<<(E + 255) / 256, blk, 0, stream>>>(dstv, dinv, E);
  k_rsqrt<<<(N + 255) / 256, blk, 0, stream>>>(dinv, N);

  // layer 1: h = x @ W1e ; g = h*dinv^2 + b1e ; g[dst] += h[src]*norm
  k_gemm128_wmma<false><<<gemmBlocks, blk, 0, stream>>>(x, W1e, hbuf, N);
  k_self_bias<<<(int)((tot + 255) / 256), blk, 0, stream>>>(hbuf, dinv, b1e, gbuf, tot);
  k_edge_agg<<<edgeBlocks, blk, 0, stream>>>(hbuf, src, dstv, dinv, gbuf, E);

  // layer 2: h = relu(g) @ W2e ; g = h*dinv^2 + b2e ; g[dst] += h[src]*norm
  k_gemm128_wmma<true><<<gemmBlocks, blk, 0, stream>>>(gbuf, W2e, hbuf, N);
  k_self_bias<<<(int)((tot + 255) / 256), blk, 0, stream>>>(hbuf, dinv, b2e, gbuf, tot);
  k_edge_agg<<<edgeBlocks, blk, 0, stream>>>(hbuf, src, dstv, dinv, gbuf, E);

  // mean pool (relu fused into read) + final linear
  k_zero<<<(G * CH + G + 255) / 256, blk, 0, stream>>>(pooled, G * CH + G);
  k_pool<<<poolBlocks, blk, 0, stream>>>(gbuf, batch, pooled, counts, N);
  k_final<<<G, 128, 0, stream>>>(pooled, counts, Wlin, blin, out, COUT);
}